// Stripe_NonLocalBlock_10376640987519
// MI455X (gfx1250) — compile-verified
//
#include <hip/hip_runtime.h>
#include <hip/hip_bf16.h>

// ---------------------------------------------------------------------------
// Stripe non-local block, MI455X (gfx1250, wave32).
//   b=4, c=256, t=8, h=128, w=64, STRIPE=8 -> hs=16, S=t*8=64 tokens, ic=128
// Memory-bound: 2x read of x (256MB) + 1x write (256MB) ~ 768MB @ 23.3TB/s.
// All GEMMs are tiny (<70 MFLOP) -> done with V_WMMA_F32_16X16X4_F32 (full
// f32 precision, native CDNA5 matrix path).
// ---------------------------------------------------------------------------

typedef __attribute__((ext_vector_type(2))) float v2f;
typedef __attribute__((ext_vector_type(8))) float v8f;

// D = A(16x4,f32) * B(4x16,f32) + C(16x16,f32), wave32.
// 8-arg form: (neg_a, A, neg_b, B, c_mod, C, reuse_a, reuse_b)
__device__ __forceinline__ v8f wmma_k4(v2f a, v2f b, v8f c) {
    return __builtin_amdgcn_wmma_f32_16x16x4_f32(
        false, a, false, b, (short)0, c, false, false);
}

#define NB   4
#define NC   256
#define NT   8
#define NH   128
#define NW   64
#define NS   64      // tokens per batch: t * STRIPE
#define NIC  128     // inter channels
#define REG  1024    // elements per stripe region (16*64), contiguous in x

// ---------------------------------------------------------------------------
// K1: stripe mean pool. One wave32 per region (1024 contiguous floats).
//     d[r] layout == (b, c, S) flattened, r = ((b*256+c)*8+t)*8+s.
// ---------------------------------------------------------------------------
__global__ void pool_kernel(const float* __restrict__ x, float* __restrict__ d,
                            int nregions) {
    int wave = threadIdx.x >> 5;
    int lane = threadIdx.x & 31;
    int r = blockIdx.x * 8 + wave;
    if (r >= nregions) return;
    const float4* p = (const float4*)(x + (size_t)r * REG);
    float s = 0.f;
#pragma unroll
    for (int j = 0; j < 8; ++j) {            // 32 lanes * 8 float4 = 1024 f32
        float4 v = p[lane + 32 * j];
        s += v.x + v.y + v.z + v.w;
    }
#pragma unroll
    for (int off = 16; off > 0; off >>= 1) s += __shfl_xor(s, off, 32);
    if (lane == 0) d[r] = s * (1.0f / (float)REG);
}

// ---------------------------------------------------------------------------
// K2: fused g / theta / phi projections.  O[b,s,i] = sum_c d[b,c,s]*W[i,c]+bias
// One 16x16 tile per wave; tiles: 3 proj * 4 batch * (4 x 8) = 384 waves.
// A = dn^T (64x256), B[k][n] = W[n][k], K-loop 256/4.
// ---------------------------------------------------------------------------
__global__ void proj_kernel(const float* __restrict__ d,
                            const float* __restrict__ gw, const float* __restrict__ gb,
                            const float* __restrict__ tw, const float* __restrict__ tb,
                            const float* __restrict__ pw, const float* __restrict__ pb,
                            float* __restrict__ g, float* __restrict__ th,
                            float* __restrict__ ph) {
    int wid  = blockIdx.x * 8 + (threadIdx.x >> 5);
    int lane = threadIdx.x & 31;
    int hi = lane >> 4, lo = lane & 15;

    int proj = wid >> 7;         // 0..2
    int rem  = wid & 127;
    int b    = rem >> 5;         // 0..3
    int tile = rem & 31;
    int tm = tile >> 3;          // 0..3 (S tiles)
    int tn = tile & 7;           // 0..7 (ic tiles)

    const float* W; const float* Bv; float* O;
    if (proj == 0)      { W = gw; Bv = gb; O = g;  }
    else if (proj == 1) { W = tw; Bv = tb; O = th; }
    else                { W = pw; Bv = pb; O = ph; }

    const float* dB = d + (size_t)b * NC * NS;   // [c][s]
    int mrow = tm * 16 + lo;                     // A row (token s)
    int ncol = tn * 16 + lo;                     // B col (out channel i)

    v8f acc = {};
    for (int k = 0; k < NC; k += 4) {
        int kk = k + 2 * hi;
        v2f a, bb;
        a.x  = dB[(kk + 0) * NS + mrow];         // A[m][kk]   = d[b][kk][m]
        a.y  = dB[(kk + 1) * NS + mrow];
        bb.x = W[(size_t)ncol * NC + kk + 0];    // B[kk][n]   = W[n][kk]
        bb.y = W[(size_t)ncol * NC + kk + 1];
        acc = wmma_k4(a, bb, acc);
    }
    float bias = Bv[ncol];
#pragma unroll
    for (int v = 0; v < 8; ++v) {
        int row = tm * 16 + v + 8 * hi;          // token s
        O[((size_t)b * NS + row) * NIC + tn * 16 + lo] = acc[v] + bias;
    }
}

// ---------------------------------------------------------------------------
// K3: per-batch attention.  f = th @ ph^T (64x64) -> softmax rows -> y = f @ g
// grid = 4 blocks of 256 threads (8 waves). f kept in LDS (16 KB).
// ---------------------------------------------------------------------------
__global__ void attn_kernel(const float* __restrict__ g,
                            const float* __restrict__ th,
                            const float* __restrict__ ph,
                            float* __restrict__ y) {
    __shared__ float f[NS][NS];
    int b    = blockIdx.x;
    int wid  = threadIdx.x >> 5;
    int lane = threadIdx.x & 31;
    int hi = lane >> 4, lo = lane & 15;
    const float* thB = th + (size_t)b * NS * NIC;
    const float* phB = ph + (size_t)b * NS * NIC;
    const float* gB  = g  + (size_t)b * NS * NIC;

    // f = th @ ph^T : 4x4 tiles, each wave does 2 (uniform per wave, EXEC full)
    for (int tt = wid; tt < 16; tt += 8) {
        int tm = tt >> 2, tn = tt & 3;
        v8f acc = {};
        for (int k = 0; k < NIC; k += 4) {
            int kk = k + 2 * hi;
            v2f a, bb;
            a.x  = thB[(tm * 16 + lo) * NIC + kk + 0];
            a.y  = thB[(tm * 16 + lo) * NIC + kk + 1];
            bb.x = phB[(tn * 16 + lo) * NIC + kk + 0];   // B[kk][n] = ph[n][kk]
            bb.y = phB[(tn * 16 + lo) * NIC + kk + 1];
            acc = wmma_k4(a, bb, acc);
        }
#pragma unroll
        for (int v = 0; v < 8; ++v)
            f[tm * 16 + v + 8 * hi][tn * 16 + lo] = acc[v];
    }
    __syncthreads();

    // softmax over each of the 64 rows (tiny)
    if (threadIdx.x < NS) {
        float* row = f[threadIdx.x];
        float m = row[0];
        for (int j = 1; j < NS; ++j) m = fmaxf(m, row[j]);
        float s = 0.f;
        for (int j = 0; j < NS; ++j) { float e = __expf(row[j] - m); row[j] = e; s += e; }
        float r = 1.0f / s;
        for (int j = 0; j < NS; ++j) row[j] *= r;
    }
    __syncthreads();

    // y = attn @ g : 4x8 tiles, each wave does 4; A gathered from LDS
    for (int tt = wid; tt < 32; tt += 8) {
        int tm = tt >> 3, tn = tt & 7;
        v8f acc = {};
        for (int k = 0; k < NS; k += 4) {
            int kk = k + 2 * hi;
            v2f a, bb;
            a.x  = f[tm * 16 + lo][kk + 0];
            a.y  = f[tm * 16 + lo][kk + 1];
            bb.x = gB[(size_t)(kk + 0) * NIC + tn * 16 + lo]; // B[kk][n] = g[kk][n]
            bb.y = gB[(size_t)(kk + 1) * NIC + tn * 16 + lo];
            acc = wmma_k4(a, bb, acc);
        }
#pragma unroll
        for (int v = 0; v < 8; ++v)
            y[((size_t)b * NS + tm * 16 + v + 8 * hi) * NIC + tn * 16 + lo] = acc[v];
    }
}

// ---------------------------------------------------------------------------
// K4: Wy[b,cc,s] = BN( sum_i y[b,s,i]*W_w[cc,i] + W_b[cc] )
// Bias+BN folded: val = acc*inv + add, inv = gamma*rsqrt(var+eps),
// add = (W_b - mean)*inv + beta. 4 batch * (4 s-tiles x 16 c-tiles) = 256 waves.
// ---------------------------------------------------------------------------
__global__ void outproj_kernel(const float* __restrict__ y,
                               const float* __restrict__ Ww, const float* __restrict__ Wb,
                               const float* __restrict__ gamma, const float* __restrict__ beta,
                               const float* __restrict__ mean,  const float* __restrict__ var,
                               float* __restrict__ Wy) {
    int wid  = blockIdx.x * 8 + (threadIdx.x >> 5);
    int lane = threadIdx.x & 31;
    int hi = lane >> 4, lo = lane & 15;
    int b    = wid >> 6;
    int tile = wid & 63;
    int tm = tile >> 4;          // s tile 0..3
    int tn = tile & 15;          // c tile 0..15
    const float* yB = y + (size_t)b * NS * NIC;
    int ncol = tn * 16 + lo;     // out channel cc

    v8f acc = {};
    for (int k = 0; k < NIC; k += 4) {
        int kk = k + 2 * hi;
        v2f a, bb;
        a.x  = yB[(tm * 16 + lo) * NIC + kk + 0];
        a.y  = yB[(tm * 16 + lo) * NIC + kk + 1];
        bb.x = Ww[(size_t)ncol * NIC + kk + 0];          // B[kk][n] = W_w[n][kk]
        bb.y = Ww[(size_t)ncol * NIC + kk + 1];
        acc = wmma_k4(a, bb, acc);
    }
    float inv = gamma[ncol] * rsqrtf(var[ncol] + 1e-5f);
    float add = (Wb[ncol] - mean[ncol]) * inv + beta[ncol];
#pragma unroll
    for (int v = 0; v < 8; ++v) {
        int row = tm * 16 + v + 8 * hi;                  // token s
        Wy[((size_t)b * NC + ncol) * NS + row] = acc[v] * inv + add;
    }
}

// ---------------------------------------------------------------------------
// K5: z = x + Wy[region], float4 streaming (the bandwidth-dominant pass).
// Each region is 1024 contiguous floats; Wy flat index == region index.
// ---------------------------------------------------------------------------
__global__ void bcast_add_kernel(const float* __restrict__ x,
                                 const float* __restrict__ Wy,
                                 float* __restrict__ out) {
    size_t i = (size_t)blockIdx.x * blockDim.x + threadIdx.x;  // float4 index
    float4 v = ((const float4*)x)[i];
    float a = Wy[i >> 8];                                      // (i*4)/1024
    float4 o = { v.x + a, v.y + a, v.z + a, v.w + a };
    ((float4*)out)[i] = o;
}

// ---------------------------------------------------------------------------
extern "C" void kernel_launch(void* const* d_in, const int* in_sizes, int n_in,
                              void* d_out, int out_size, void* d_ws, size_t ws_size,
                              hipStream_t stream) {
    const float* x     = (const float*)d_in[0];
    const float* g_w   = (const float*)d_in[1];
    const float* g_b   = (const float*)d_in[2];
    const float* th_w  = (const float*)d_in[3];
    const float* th_b  = (const float*)d_in[4];
    const float* ph_w  = (const float*)d_in[5];
    const float* ph_b  = (const float*)d_in[6];
    const float* W_w   = (const float*)d_in[7];
    const float* W_b   = (const float*)d_in[8];
    const float* bn_g  = (const float*)d_in[9];
    const float* bn_b  = (const float*)d_in[10];
    const float* bn_m  = (const float*)d_in[11];
    const float* bn_v  = (const float*)d_in[12];

    float* ws = (float*)d_ws;              // ~1 MB of f32 scratch used
    float* d   = ws;                       // (b,c,S)        65536
    float* g   = ws + 65536;               // (b,S,ic)       32768
    float* th  = ws + 98304;               // (b,S,ic)       32768
    float* ph  = ws + 131072;              // (b,S,ic)       32768
    float* y   = ws + 163840;              // (b,S,ic)       32768
    float* Wy  = ws + 196608;              // (b,c,S)        65536

    const int nregions = NB * NC * NT * 8;                  // 65536
    pool_kernel<<<nregions / 8, 256, 0, stream>>>(x, d, nregions);
    proj_kernel<<<48, 256, 0, stream>>>(d, g_w, g_b, th_w, th_b, ph_w, ph_b,
                                        g, th, ph);
    attn_kernel<<<NB, 256, 0, stream>>>(g, th, ph, y);
    outproj_kernel<<<32, 256, 0, stream>>>(y, W_w, W_b, bn_g, bn_b, bn_m, bn_v, Wy);

    const size_t n4 = (size_t)NB * NC * NT * NH * NW / 4;   // 16,777,216 float4
    bcast_add_kernel<<<(unsigned)(n4 / 256), 256, 0, stream>>>(x, Wy, (float*)d_out);
}